// GATLayer_19688130085110
// MI455X (gfx1250) — compile-verified
//
#include <hip/hip_runtime.h>

#define HEADS   4
#define OUT_DIM 64
#define IN_DIM  256
#define HC      (HEADS * OUT_DIM)   // 256
#define ALPHA   0.2f

typedef __attribute__((ext_vector_type(2))) float v2f;
typedef __attribute__((ext_vector_type(8))) float v8f;

// Monotonic order-preserving float<->uint mapping for atomicMax on floats.
__device__ __forceinline__ unsigned enc_f32(float f) {
  unsigned u = __float_as_uint(f);
  return (u & 0x80000000u) ? ~u : (u | 0x80000000u);
}
__device__ __forceinline__ float dec_f32(unsigned u) {
  return (u & 0x80000000u) ? __uint_as_float(u & 0x7fffffffu)
                           : __uint_as_float(~u);
}

// ---------------------------------------------------------------------------
// Kernel 1: ft = feat @ W^T  via V_WMMA_F32_16X16X4_F32 (fp32 tensor path).
// grid.x = n_nodes/16 (exact: 50000 = 3125*16); block = 256 threads = 8 waves.
// Wave w computes the 16(M) x 32(N) slab at columns [32w, 32w+32).
// ---------------------------------------------------------------------------
__global__ void __launch_bounds__(256)
gat_proj_wmma(const float* __restrict__ feat, const float* __restrict__ W,
              float* __restrict__ ft) {
  const int lane  = threadIdx.x & 31;
  const int wave  = threadIdx.x >> 5;
  const int mBase = blockIdx.x << 4;
  const int nBase = wave << 5;
  const int r16   = lane & 15;   // M row (A) / N col (B) within tile
  const int hi    = lane >> 4;   // K-pair selector for frags, M-half for D

  const float* arow  = feat + (size_t)(mBase + r16) * IN_DIM;
  const float* b0row = W + (size_t)(nBase + r16) * IN_DIM;        // B[k][n]=W[n][k]
  const float* b1row = W + (size_t)(nBase + 16 + r16) * IN_DIM;

  v8f acc0 = {};
  v8f acc1 = {};
  for (int k = 0; k < IN_DIM; k += 4) {
    const int kk = k + (hi << 1);
    v2f a  = { arow[kk],  arow[kk + 1]  };   // A 16x4: v0=K(0|2), v1=K(1|3)
    v2f b0 = { b0row[kk], b0row[kk + 1] };
    v2f b1 = { b1row[kk], b1row[kk + 1] };
    acc0 = __builtin_amdgcn_wmma_f32_16x16x4_f32(false, a, false, b0,
                                                 (short)0, acc0, false, false);
    acc1 = __builtin_amdgcn_wmma_f32_16x16x4_f32(false, a, false, b1,
                                                 (short)0, acc1, false, false);
  }

  // D layout: VGPR r -> row r + 8*hi, col = lane&15 (acc1 at col+16).
  float* drow = ft + (size_t)(mBase + (hi << 3)) * HC + nBase + r16;
#pragma unroll
  for (int r = 0; r < 8; ++r) {
    drow[(size_t)r * HC]      = acc0[r];
    drow[(size_t)r * HC + 16] = acc1[r];
  }
}

// ---------------------------------------------------------------------------
// Kernel 2: per-(node,head) attention terms a1/a2 + init of max/sum arrays.
// ---------------------------------------------------------------------------
__global__ void gat_attn_init(const float* __restrict__ ft,
                              const float* __restrict__ attn_l,
                              const float* __restrict__ attn_r,
                              float* __restrict__ a1, float* __restrict__ a2,
                              unsigned* __restrict__ menc,
                              float* __restrict__ ssum, int nh) {
  const int i = blockIdx.x * blockDim.x + threadIdx.x;
  if (i >= nh) return;
  const int h = i & (HEADS - 1);
  const float* f  = ft + (size_t)i * OUT_DIM;  // [n][h][d] contiguous
  const float* al = attn_l + h * OUT_DIM;
  const float* ar = attn_r + h * OUT_DIM;
  float s1 = 0.f, s2 = 0.f;
#pragma unroll 8
  for (int d = 0; d < OUT_DIM; ++d) {
    const float v = f[d];
    s1 += v * al[d];
    s2 += v * ar[d];
  }
  a1[i] = s1;
  a2[i] = s2;
  menc[i] = 0u;     // below every real enc_f32 value
  ssum[i] = 0.f;
}

__global__ void gat_zero(float* __restrict__ p, int n) {
  const int i = blockIdx.x * blockDim.x + threadIdx.x;
  if (i < n) p[i] = 0.f;
}

// ---------------------------------------------------------------------------
// Kernel 3: segment max of leaky_relu edge scores (per dst,head).
// ---------------------------------------------------------------------------
__global__ void gat_edge_max(const int* __restrict__ src, const int* __restrict__ dst,
                             const float* __restrict__ a1, const float* __restrict__ a2,
                             unsigned* __restrict__ menc, int neh) {
  const int i = blockIdx.x * blockDim.x + threadIdx.x;
  if (i >= neh) return;
  const int e = i >> 2, h = i & 3;
  const int s = src[e], d = dst[e];
  float x = a1[s * HEADS + h] + a2[d * HEADS + h];
  x = x > 0.f ? x : ALPHA * x;
  atomicMax(&menc[d * HEADS + h], enc_f32(x));
}

// ---------------------------------------------------------------------------
// Kernel 4: segment sum of exp(e - max).
// ---------------------------------------------------------------------------
__global__ void gat_edge_sum(const int* __restrict__ src, const int* __restrict__ dst,
                             const float* __restrict__ a1, const float* __restrict__ a2,
                             const unsigned* __restrict__ menc,
                             float* __restrict__ ssum, int neh) {
  const int i = blockIdx.x * blockDim.x + threadIdx.x;
  if (i >= neh) return;
  const int e = i >> 2, h = i & 3;
  const int s = src[e], d = dst[e];
  float x = a1[s * HEADS + h] + a2[d * HEADS + h];
  x = x > 0.f ? x : ALPHA * x;
  atomicAdd(&ssum[d * HEADS + h], __expf(x - dec_f32(menc[d * HEADS + h])));
}

// ---------------------------------------------------------------------------
// Kernel 5: weighted scatter-aggregate. One edge per 256-thread block:
// thread c handles channel c (head = c/64). Coalesced gather of ft[src],
// coalesced float atomic-add scatter into out[dst].
// ---------------------------------------------------------------------------
__global__ void __launch_bounds__(256)
gat_edge_aggr(const int* __restrict__ src, const int* __restrict__ dst,
              const float* __restrict__ a1, const float* __restrict__ a2,
              const unsigned* __restrict__ menc, const float* __restrict__ ssum,
              const float* __restrict__ ft, float* __restrict__ out) {
  const int e = blockIdx.x;
  const int c = threadIdx.x;       // 0..255
  const int h = c >> 6;
  const int s = src[e], d = dst[e];
  float x = a1[s * HEADS + h] + a2[d * HEADS + h];
  x = x > 0.f ? x : ALPHA * x;
  const float a = __expf(x - dec_f32(menc[d * HEADS + h])) / ssum[d * HEADS + h];
  atomicAdd(&out[(size_t)d * HC + c], a * ft[(size_t)s * HC + c]);
}

// ---------------------------------------------------------------------------
// Kernel 6: in-place ELU on the accumulated output.
// ---------------------------------------------------------------------------
__global__ void gat_elu(float* __restrict__ out, int n) {
  const int i = blockIdx.x * blockDim.x + threadIdx.x;
  if (i < n) {
    const float x = out[i];
    out[i] = x > 0.f ? x : __expf(x) - 1.f;
  }
}

extern "C" void kernel_launch(void* const* d_in, const int* in_sizes, int n_in,
                              void* d_out, int out_size, void* d_ws, size_t ws_size,
                              hipStream_t stream) {
  const float* feat   = (const float*)d_in[0];
  const int*   src    = (const int*)d_in[1];
  const int*   dst    = (const int*)d_in[2];
  const float* W      = (const float*)d_in[3];
  const float* attn_l = (const float*)d_in[4];
  const float* attn_r = (const float*)d_in[5];
  float* out = (float*)d_out;

  const int n_nodes = in_sizes[0] / IN_DIM;   // 50000
  const int n_edges = in_sizes[1];            // 800000
  const int nh  = n_nodes * HEADS;            // 200000
  const int nc  = n_nodes * HC;               // 12.8M output elems
  const int neh = n_edges * HEADS;            // 3.2M

  // Workspace layout (~54.5 MB): ft | a1 | a2 | menc | ssum
  char* ws = (char*)d_ws;
  float*    ft   = (float*)ws;    ws += (size_t)n_nodes * HC * sizeof(float);
  float*    a1   = (float*)ws;    ws += (size_t)nh * sizeof(float);
  float*    a2   = (float*)ws;    ws += (size_t)nh * sizeof(float);
  unsigned* menc = (unsigned*)ws; ws += (size_t)nh * sizeof(unsigned);
  float*    ssum = (float*)ws;

  gat_proj_wmma<<<n_nodes / 16, 256, 0, stream>>>(feat, W, ft);
  gat_zero<<<(nc + 255) / 256, 256, 0, stream>>>(out, nc);
  gat_attn_init<<<(nh + 255) / 256, 256, 0, stream>>>(ft, attn_l, attn_r,
                                                      a1, a2, menc, ssum, nh);
  gat_edge_max<<<(neh + 255) / 256, 256, 0, stream>>>(src, dst, a1, a2, menc, neh);
  gat_edge_sum<<<(neh + 255) / 256, 256, 0, stream>>>(src, dst, a1, a2, menc, ssum, neh);
  gat_edge_aggr<<<n_edges, 256, 0, stream>>>(src, dst, a1, a2, menc, ssum, ft, out);
  gat_elu<<<(nc + 255) / 256, 256, 0, stream>>>(out, nc);
}